// LinearList_69861938037355
// MI455X (gfx1250) — compile-verified
//
#include <hip/hip_runtime.h>

// ---- problem constants (match reference) ----
#define N_TOK  8192
#define DIN    1024
#define DOUT   1024
#define NEXP   8

// ---- tiling ----
#define BLOCK_M   128                // tokens per workgroup tile
#define KCHUNK    64                 // K staged in LDS per iteration
#define LDSSTRIDE 68                 // 64 + 4 pad -> conflict-free ds_load_b64
#define BUF_FLOATS (BLOCK_M * LDSSTRIDE)
#define GRID_M    ((N_TOK + NEXP*(BLOCK_M-1) + BLOCK_M - 1) / BLOCK_M)  // 72
#define ORDER_CAP (GRID_M * BLOCK_M) // 9216
#define COLS_PER_WG   256            // blockIdx.y in [0,4)
#define COLS_PER_WAVE 32             // 2 x 16-col subtiles per wave

typedef float v2f __attribute__((ext_vector_type(2)));
typedef float v8f __attribute__((ext_vector_type(8)));

// ws int layout: [0..7] counts, [8..15] cursors, [16..23] offsets,
// [32..32+GRID_M) tile_expert, [256..256+ORDER_CAP) order
#define WS_COUNTS   0
#define WS_CURSORS  8
#define WS_OFFS     16
#define WS_TILEEXP  32
#define WS_ORDER    256

__global__ void init_kernel(int* ws) {
    int i = blockIdx.x * blockDim.x + threadIdx.x;
    if (i < ORDER_CAP) ws[WS_ORDER + i] = -1;
    if (blockIdx.x == 0 && threadIdx.x < 16) ws[threadIdx.x] = 0; // counts+cursors
}

__global__ void count_kernel(const int* __restrict__ types, int* ws) {
    int i = blockIdx.x * blockDim.x + threadIdx.x;
    if (i < N_TOK) atomicAdd(&ws[WS_COUNTS + types[i]], 1);
}

__global__ void scan_kernel(int* ws) {
    if (threadIdx.x != 0) return;
    for (int t = 0; t < GRID_M; ++t) ws[WS_TILEEXP + t] = -1;
    int running = 0;
    for (int e = 0; e < NEXP; ++e) {
        ws[WS_OFFS + e] = running;
        int tiles = (ws[WS_COUNTS + e] + BLOCK_M - 1) / BLOCK_M;
        int t0 = running / BLOCK_M;
        for (int t = 0; t < tiles; ++t) ws[WS_TILEEXP + t0 + t] = e;
        running += tiles * BLOCK_M;
    }
}

__global__ void scatter_kernel(const int* __restrict__ types, int* ws) {
    int i = blockIdx.x * blockDim.x + threadIdx.x;
    if (i < N_TOK) {
        int e = types[i];
        int pos = ws[WS_OFFS + e] + atomicAdd(&ws[WS_CURSORS + e], 1);
        ws[WS_ORDER + pos] = i;
    }
}

// D = X_tile @ W_e^T + b_e using V_WMMA_F32_16X16X4_F32,
// X staged through LDS with async-to-LDS double buffering.
__global__ __launch_bounds__(256) void gemm_kernel(
    const float* __restrict__ X, const float* __restrict__ W,
    const float* __restrict__ Bias, const int* __restrict__ ws,
    float* __restrict__ Out)
{
    __shared__ float xs[2 * BUF_FLOATS];

    const int tile = blockIdx.x;
    const int e = ws[WS_TILEEXP + tile];
    if (e < 0) return;  // uniform over the whole block

    const int tid  = threadIdx.x;
    const int lane = tid & 31;
    const int wave = tid >> 5;
    const int half = lane >> 4;   // 0: lanes 0-15, 1: lanes 16-31
    const int l16  = lane & 15;
    const int col_base = blockIdx.y * COLS_PER_WG + wave * COLS_PER_WAVE;
    const int* order = &ws[WS_ORDER + tile * BLOCK_M];
    const float* We = W + (size_t)e * DOUT * DIN;

    // ---- per-thread gather descriptors for async X staging ----
    // 128 rows x 16 float4 = 2048 b128 slots, 8 per thread.
    // Filler rows (order<0) are clamped to row 0: garbage only affects
    // discarded output rows (matmul rows are independent), never stored.
    const unsigned lds_base = (unsigned)(size_t)(&xs[0]);
    const float* gptr0[8];
    unsigned ldsoff0[8];
#pragma unroll
    for (int i = 0; i < 8; ++i) {
        int s   = tid + i * 256;
        int row = s >> 4;
        int c4  = s & 15;
        int g   = order[row];
        if (g < 0) g = 0;
        gptr0[i]  = X + (size_t)g * DIN + c4 * 4;
        ldsoff0[i] = lds_base + (unsigned)(row * LDSSTRIDE + c4 * 4) * 4u;
    }

    v8f acc[8][2];
#pragma unroll
    for (int ms = 0; ms < 8; ++ms)
#pragma unroll
        for (int ns = 0; ns < 2; ++ns)
            acc[ms][ns] = (v8f)(0.0f);

    // prologue: async-stage chunk 0 into buffer 0
#pragma unroll
    for (int i = 0; i < 8; ++i) {
        asm volatile("global_load_async_to_lds_b128 %0, %1, off"
                     :: "v"(ldsoff0[i]), "v"(gptr0[i]) : "memory");
    }
    asm volatile("s_wait_asynccnt 0x0" ::: "memory");
    __syncthreads();

    int cur = 0;
    for (int kc = 0; kc < DIN; kc += KCHUNK) {
        // prefetch next chunk into the other buffer (async engine overlaps WMMA)
        if (kc + KCHUNK < DIN) {
            unsigned boff = (unsigned)((cur ^ 1) * BUF_FLOATS) * 4u;
#pragma unroll
            for (int i = 0; i < 8; ++i) {
                const float* gp = gptr0[i] + kc + KCHUNK;
                unsigned la = ldsoff0[i] + boff;
                asm volatile("global_load_async_to_lds_b128 %0, %1, off"
                             :: "v"(la), "v"(gp) : "memory");
            }
        }

        const float* xb = &xs[cur * BUF_FLOATS];
#pragma unroll 2
        for (int kk = 0; kk < KCHUNK; kk += 4) {
            v2f afrag[8];
#pragma unroll
            for (int ms = 0; ms < 8; ++ms)
                afrag[ms] = *(const v2f*)(&xb[(ms * 16 + l16) * LDSSTRIDE + kk + 2 * half]);
#pragma unroll
            for (int ns = 0; ns < 2; ++ns) {
                int col = col_base + ns * 16 + l16;       // output column = W row
                v2f bfrag = *(const v2f*)(We + (size_t)col * DIN + kc + kk + 2 * half);
#pragma unroll
                for (int ms = 0; ms < 8; ++ms)
                    acc[ms][ns] = __builtin_amdgcn_wmma_f32_16x16x4_f32(
                        false, afrag[ms], false, bfrag,
                        (short)0, acc[ms][ns], false, false);
            }
        }

        __syncthreads();                                   // all waves done reading buf[cur]
        asm volatile("s_wait_asynccnt 0x0" ::: "memory");  // my prefetch landed in LDS
        __syncthreads();                                   // everyone's prefetch landed
        cur ^= 1;
    }

    // epilogue: add bias, scatter rows back to token order
#pragma unroll
    for (int ns = 0; ns < 2; ++ns) {
        int col = col_base + ns * 16 + l16;
        float bv = Bias[e * DOUT + col];
#pragma unroll
        for (int ms = 0; ms < 8; ++ms) {
#pragma unroll
            for (int v = 0; v < 8; ++v) {
                int row = ms * 16 + v + 8 * half;   // C/D layout: VGPR v -> M = v + 8*half
                int g = order[row];
                if (g >= 0)
                    Out[(size_t)g * DOUT + col] = acc[ms][ns][v] + bv;
            }
        }
    }
}

extern "C" void kernel_launch(void* const* d_in, const int* in_sizes, int n_in,
                              void* d_out, int out_size, void* d_ws, size_t ws_size,
                              hipStream_t stream) {
    const float* X    = (const float*)d_in[0];   // [8192,1024]
    const float* W    = (const float*)d_in[1];   // [8,1024,1024]
    const float* Bias = (const float*)d_in[2];   // [8,1024]
    const int*   T    = (const int*)d_in[3];     // [8192]
    float* Out = (float*)d_out;
    int* ws = (int*)d_ws;

    init_kernel<<<(ORDER_CAP + 255) / 256, 256, 0, stream>>>(ws);
    count_kernel<<<N_TOK / 256, 256, 0, stream>>>(T, ws);
    scan_kernel<<<1, 32, 0, stream>>>(ws);
    scatter_kernel<<<N_TOK / 256, 256, 0, stream>>>(T, ws);

    dim3 grid(GRID_M, DOUT / COLS_PER_WG);  // (72, 4)
    gemm_kernel<<<grid, 256, 0, stream>>>(X, W, Bias, ws, Out);
}